// decoder_66752381714625
// MI455X (gfx1250) — compile-verified
//
#include <hip/hip_runtime.h>

// ---------------------------------------------------------------------------
// Weighted BP LDPC decoder for MI455X (gfx1250).
//
// Shapes: M=512 checks, N=1024 vars, DV=3, DC=6, ITERS=10, E=3072, B=8192.
// K=2 batch items per 256-thread block: graph indices (registers) and
// per-iteration weights (L2) are loaded ONCE per block and applied to both
// items, halving L2 weight/index bandwidth and doubling ILP through the
// transcendental pipe. All message state lives in LDS (64 KB/block ->
// 5 blocks per 320 KB WGP). Both llr rows (contiguous in memory) arrive via
// a single Tensor Data Mover DMA (tensor_load_to_lds, TENSORcnt).
// ---------------------------------------------------------------------------

#define NN     1024
#define ITERS  10
#define EE     3072
#define TWW    6144
#define NOUT   512
#define TPB    256
#define KB     2               // batch items per block
#define EPT    (EE / TPB)      // 12 edges per thread
#define VPT    (NN / TPB)      // 4 variables per thread

typedef unsigned int u32;
typedef __attribute__((ext_vector_type(4))) u32 tdm_v4u;
typedef __attribute__((ext_vector_type(8))) int tdm_v8i;
typedef __attribute__((ext_vector_type(4))) int tdm_v4i;

__device__ __forceinline__ float fexp2(float x) { return __builtin_amdgcn_exp2f(x); }
__device__ __forceinline__ float flog2(float x) { return __builtin_amdgcn_logf(x); }
__device__ __forceinline__ float frcp (float x) { return __builtin_amdgcn_rcpf(x); }

#define LOG2E 1.44269504088896340736f
#define LN2   0.69314718055994530942f
#define CLIP  0.999995f

__global__ __launch_bounds__(TPB) void bp_decode_kernel(
    const float* __restrict__ llr,        // [B, N]
    const float* __restrict__ w_iter,     // [ITERS, E*(DV-1)]
    const float* __restrict__ llr_iter,   // [ITERS, N]
    const float* __restrict__ w_final,    // [E]  (indexed by flat (n,k) position)
    const float* __restrict__ llr_final,  // [N]
    const int*   __restrict__ v_sum_idx,  // [E, 2]
    const int*   __restrict__ c_prod_idx, // [E, 5]
    const int*   __restrict__ edge_var,   // [E]
    const int*   __restrict__ final_idx,  // [N, 3]
    float*       __restrict__ out)        // [B, NOUT]
{
    __shared__ __align__(16) float s_llr[KB][NN];   // llr rows (filled by one TDM DMA)
    __shared__ __align__(16) float s_wl [KB][NN];   // llr * llr_iter (per iteration)
    __shared__ __align__(16) float s_v2c[KB][EE];   // variable->check messages
    __shared__ __align__(16) float s_c2v[KB][EE];   // check->variable messages

    const int tid = threadIdx.x;
    const int b0  = blockIdx.x * KB;                // first batch item of this block

    // ---- Single TDM load of KB contiguous llr rows (8 KB) into s_llr ----
    // Wave 0 only (TDM ignores EXEC; one issue per wave executing it).
    if (tid < 32) {
        const float* gsrc = llr + (size_t)b0 * NN;
        unsigned long long ga = (unsigned long long)(uintptr_t)gsrc;
        u32 lds_addr = (u32)(uintptr_t)(void*)&s_llr[0][0];  // low 32 bits = LDS offset
        const u32 nel = KB * NN;                             // 2048 contiguous f32

        // D# group 0 (128b): count=1, lds_addr, global_addr[56:0], type=2
        tdm_v4u g0;
        g0.x = 1u;                                    // count=1 (valid user descriptor)
        g0.y = lds_addr;                              // lds_addr [63:32]
        g0.z = (u32)(ga & 0xffffffffu);               // global_addr[31:0]
        g0.w = (u32)((ga >> 32) & 0x01ffffffu)        // global_addr[56:32]
             | 0x80000000u;                           // type=2 ("image") [127:126]

        // D# group 1 (256b): data_size=4B, 2048x1 tensor, 2048x1 tile
        tdm_v8i g1;
        g1[0] = 0x00020000;                           // data_size=2 (4B); wg_mask=0
        g1[1] = (int)((nel & 0xffffu) << 16);         // tensor_dim0[15:0] in [63:48]
        g1[2] = 0x00010000;                           // tensor_dim0[31:16]=0 | tensor_dim1=1
        g1[3] = (int)((nel & 0xffffu) << 16);         // tile_dim0=2048 in [127:112]
        g1[4] = 0x00000001;                           // tile_dim1=1, tile_dim2=0
        g1[5] = (int)nel;                             // tensor_dim0_stride lo
        g1[6] = 0;                                    // stride0 hi | stride1 lo
        g1[7] = 0;                                    // stride1 hi

        tdm_v4i g2 = {0, 0, 0, 0};                    // dims 2/3 unused
        tdm_v4i g3 = {0, 0, 0, 0};                    // dim 4 unused
        tdm_v8i g4 = {0, 0, 0, 0, 0, 0, 0, 0};        // extra group (clang-23 6-arg form)

        __builtin_amdgcn_tensor_load_to_lds(g0, g1, g2, g3, g4, 0);
    }

    // ---- While the DMA is in flight: preload graph indices into registers
    //      (loaded once, reused for all iterations AND both batch items). ----
    int va[EPT], vb[EPT], ev[EPT];
    int cp[EPT][5];
#pragma unroll
    for (int i = 0; i < EPT; ++i) {
        const int e = tid + TPB * i;
        va[i] = v_sum_idx[2 * e];
        vb[i] = v_sum_idx[2 * e + 1];
        ev[i] = edge_var[e];
#pragma unroll
        for (int j = 0; j < 5; ++j) cp[i][j] = c_prod_idx[5 * e + j];
#pragma unroll
        for (int k = 0; k < KB; ++k) s_c2v[k][e] = 0.0f;
    }

    // Wave 0 waits for the tensor DMA, then the whole block synchronizes.
    if (tid < 32) {
        __builtin_amdgcn_s_wait_tensorcnt(0);
    }
    __syncthreads();

    // ---- BP iterations ----
    for (int it = 0; it < ITERS; ++it) {
        // Per-variable weighted channel LLR (llr_iter row loaded once/block).
        const float* __restrict__ lwp = llr_iter + it * NN;
#pragma unroll
        for (int i = 0; i < VPT; ++i) {
            const int n = tid + TPB * i;
            const float lw = lwp[n];
#pragma unroll
            for (int k = 0; k < KB; ++k) s_wl[k][n] = s_llr[k][n] * lw;
        }
        __syncthreads();

        // Phase A: variable -> check.  v2c = tanh(0.5*(w0*c2v[a] + w1*c2v[b] + wl))
        const float* __restrict__ wp = w_iter + it * TWW;
#pragma unroll
        for (int i = 0; i < EPT; ++i) {
            const int e = tid + TPB * i;
            const float w0 = wp[2 * e];               // loaded once, used for KB items
            const float w1 = wp[2 * e + 1];
#pragma unroll
            for (int k = 0; k < KB; ++k) {
                const float s = s_c2v[k][va[i]] * w0 + s_c2v[k][vb[i]] * w1
                              + s_wl[k][ev[i]];
                // tanh(s/2) = 1 - 2/(exp(s)+1)   (stable for large |s|)
                const float ex = fexp2(s * LOG2E);
                s_v2c[k][e] = 1.0f - 2.0f * frcp(ex + 1.0f);
            }
        }
        __syncthreads();

        // Phase B: check -> variable.  c2v = 2*atanh(clip * prod of 5 extrinsic v2c)
#pragma unroll
        for (int i = 0; i < EPT; ++i) {
            const int e = tid + TPB * i;
#pragma unroll
            for (int k = 0; k < KB; ++k) {
                float p = s_v2c[k][cp[i][0]] * s_v2c[k][cp[i][1]];
                p *= s_v2c[k][cp[i][2]];
                p *= s_v2c[k][cp[i][3]];
                p *= s_v2c[k][cp[i][4]];
                const float c = CLIP * p;             // |c| < 1 guaranteed
                // 2*atanh(c) = ln((1+c)/(1-c)) = ln2*(log2(1+c) - log2(1-c))
                s_c2v[k][e] = LN2 * (flog2(1.0f + c) - flog2(1.0f - c));
            }
        }
        __syncthreads();
    }

    // ---- Final marginalization + sigmoid for variables [0, NOUT) ----
    // NOTE: reference multiplies by w_final at the FLAT (n,k) position.
    float* __restrict__ orow = out + (size_t)b0 * NOUT;
#pragma unroll
    for (int i = 0; i < NOUT / TPB; ++i) {
        const int n   = tid + TPB * i;
        const int f0  = final_idx[3 * n];
        const int f1  = final_idx[3 * n + 1];
        const int f2  = final_idx[3 * n + 2];
        const float wf0 = w_final[3 * n];
        const float wf1 = w_final[3 * n + 1];
        const float wf2 = w_final[3 * n + 2];
        const float lf  = llr_final[n];
#pragma unroll
        for (int k = 0; k < KB; ++k) {
            const float fin = s_c2v[k][f0] * wf0 + s_c2v[k][f1] * wf1
                            + s_c2v[k][f2] * wf2;
            const float o = fin + s_llr[k][n] * lf;
            orow[(size_t)k * NOUT + n] = frcp(1.0f + fexp2(-o * LOG2E));  // sigmoid
        }
    }
}

extern "C" void kernel_launch(void* const* d_in, const int* in_sizes, int n_in,
                              void* d_out, int out_size, void* d_ws, size_t ws_size,
                              hipStream_t stream) {
    const float* llr       = (const float*)d_in[0];
    const float* w_iter    = (const float*)d_in[1];
    const float* llr_iter  = (const float*)d_in[2];
    const float* w_final   = (const float*)d_in[3];
    const float* llr_final = (const float*)d_in[4];
    const int*   v_sum_idx = (const int*)d_in[5];
    const int*   c_prod    = (const int*)d_in[6];
    const int*   edge_var  = (const int*)d_in[7];
    const int*   final_idx = (const int*)d_in[8];
    float*       out       = (float*)d_out;

    const int B = in_sizes[0] / NN;   // 8192
    bp_decode_kernel<<<B / KB, TPB, 0, stream>>>(llr, w_iter, llr_iter, w_final,
                                                 llr_final, v_sum_idx, c_prod,
                                                 edge_var, final_idx, out);
}